// Pocket_GNN_53429393162939
// MI455X (gfx1250) — compile-verified
//
#include <hip/hip_runtime.h>
#include <hip/hip_bf16.h>

// ---------------------------------------------------------------------------
// Types for the f32 WMMA path (V_WMMA_F32_16X16X4_F32):
//   A 16x4  -> 2 VGPRs/lane : lanes0-15 hold M=lane {K=k,k+1}, lanes16-31 {K=k+2,k+3}
//   B 4x16  -> 2 VGPRs/lane : v.x = row (k+half*2), v.y = row (k+half*2+1), col = lane&15
//   C 16x16 -> 8 VGPRs/lane : VGPR r = row (r + half*8), col = lane&15
// ---------------------------------------------------------------------------
typedef float v2f __attribute__((ext_vector_type(2)));
typedef float v8f __attribute__((ext_vector_type(8)));

__device__ __forceinline__ float eluf(float v) { return v > 0.f ? v : expm1f(v); }

// order-preserving unsigned encoding of float for atomicMax
__device__ __forceinline__ unsigned encF(float f) {
  unsigned u = __float_as_uint(f);
  return (u & 0x80000000u) ? ~u : (u | 0x80000000u);
}
__device__ __forceinline__ float decF(unsigned u) {
  return __uint_as_float((u & 0x80000000u) ? (u & 0x7fffffffu) : ~u);
}

// CDNA5 async global->LDS copy of one dword (GLOBAL_LOAD_ASYNC_TO_LDS_B32,
// GVS mode: saddr base + 32-bit voffset). ldsOff is the LDS byte address
// (low 32 bits of a flat shared pointer), tracked by ASYNCcnt.
__device__ __forceinline__ void async_copy_b32(unsigned ldsOff, unsigned gByteOff,
                                               unsigned long long gBase) {
  asm volatile("global_load_async_to_lds_b32 %0, %1, %2"
               :: "v"(ldsOff), "v"(gByteOff), "s"(gBase)
               : "memory");
}
__device__ __forceinline__ void wait_async0() {
  asm volatile("s_wait_asynccnt 0x0" ::: "memory");
}

// ---------------------------------------------------------------------------
// Pack 4 weight matrices [K,D] (q|k|v|s) into fused [Kpad, 4D] (zero pad rows)
// and 4 biases [D] into [4D].
// ---------------------------------------------------------------------------
__global__ void pack_w4(const float* __restrict__ wq, const float* __restrict__ wk,
                        const float* __restrict__ wv, const float* __restrict__ ws,
                        const float* __restrict__ bq, const float* __restrict__ bk,
                        const float* __restrict__ bv, const float* __restrict__ bs,
                        float* __restrict__ Wp, float* __restrict__ bp,
                        int K, int Kpad, int D) {
  int total = Kpad * 4 * D;
  int i = blockIdx.x * blockDim.x + threadIdx.x;
  if (i < total) {
    int k = i / (4 * D);
    int col = i - k * (4 * D);
    int sec = col / D;
    int c = col - sec * D;
    const float* w = (sec == 0) ? wq : (sec == 1) ? wk : (sec == 2) ? wv : ws;
    Wp[i] = (k < K) ? w[k * D + c] : 0.0f;
  } else if (i < total + 4 * D) {
    int col = i - total;
    int sec = col / D;
    int c = col - sec * D;
    const float* b = (sec == 0) ? bq : (sec == 1) ? bk : (sec == 2) ? bv : bs;
    bp[col] = b[c];
  }
}

// ---------------------------------------------------------------------------
// Node MLP: h0 = elu(elu(x@A+bA)@B+bB), x [N,23] -> h0 [N,24] (col23 = 0 pad)
// ---------------------------------------------------------------------------
__global__ void mlp_node(const float* __restrict__ x,
                         const float* __restrict__ Aw, const float* __restrict__ Ab,
                         const float* __restrict__ Bw, const float* __restrict__ Bb,
                         float* __restrict__ h0, int N) {
  __shared__ float sA[23 * 23], sAb[23], sB[23 * 23], sBb[23];
  for (int i = threadIdx.x; i < 23 * 23; i += blockDim.x) { sA[i] = Aw[i]; sB[i] = Bw[i]; }
  for (int i = threadIdx.x; i < 23; i += blockDim.x) { sAb[i] = Ab[i]; sBb[i] = Bb[i]; }
  __syncthreads();
  int n = blockIdx.x * blockDim.x + threadIdx.x;
  if (n >= N) return;
  float xi[23], t1[23];
#pragma unroll
  for (int j = 0; j < 23; ++j) xi[j] = x[(size_t)n * 23 + j];
  for (int o = 0; o < 23; ++o) {
    float acc = sAb[o];
#pragma unroll
    for (int j = 0; j < 23; ++j) acc += xi[j] * sA[j * 23 + o];
    t1[o] = eluf(acc);
  }
  for (int o = 0; o < 23; ++o) {
    float acc = sBb[o];
#pragma unroll
    for (int j = 0; j < 23; ++j) acc += t1[j] * sB[j * 23 + o];
    h0[(size_t)n * 24 + o] = eluf(acc);
  }
  h0[(size_t)n * 24 + 23] = 0.0f;
}

// ---------------------------------------------------------------------------
// WMMA f32 GEMM: C[M,Ncols] = A[M,K](lda) @ W[K,Ncols] + bias
// Block = 8 waves = 8 M-tiles x ONE 16-col N-tile. The block's [K,16] weight
// panel is staged into LDS once with GLOBAL_LOAD_ASYNC_TO_LDS_B32 (ASYNCcnt),
// then each wave runs the K loop with V_WMMA_F32_16X16X4_F32, B from LDS.
// ---------------------------------------------------------------------------
__global__ void wmma_gemm_f32(const float* __restrict__ A, int lda,
                              const float* __restrict__ W,
                              const float* __restrict__ bias,
                              float* __restrict__ C, int ldc,
                              int M, int K, int Ncols) {
  __shared__ float sW[192 * 16];                 // K <= 192
  int ntiles = Ncols >> 4;
  int mtiles = (M + 15) >> 4;
  int nt = blockIdx.x % ntiles;
  int mblock = blockIdx.x / ntiles;
  int col0 = nt * 16;

  // async-stage the weight panel W[0:K, col0:col0+16] -> sW[k*16+c]
  {
    unsigned long long base = (unsigned long long)W;
    for (int i = threadIdx.x; i < K * 16; i += blockDim.x) {
      int k = i >> 4, c = i & 15;
      unsigned ldsOff = (unsigned)(size_t)&sW[i];
      unsigned gOff = (unsigned)(((size_t)k * Ncols + col0 + c) * sizeof(float));
      async_copy_b32(ldsOff, gOff, base);
    }
    wait_async0();
  }
  __syncthreads();

  int wave = threadIdx.x >> 5;
  int mtile = mblock * 8 + wave;
  if (mtile < mtiles) {                          // wave-uniform: EXEC all-ones inside
    int lane = threadIdx.x & 31;
    int half = lane >> 4;                        // 0: K pair {k,k+1}, 1: {k+2,k+3}
    int row0 = mtile * 16;
    int mrow = row0 + (lane & 15);
    int mr = (mrow < M) ? mrow : (M - 1);        // clamp keeps EXEC uniform
    int col = col0 + (lane & 15);

    v8f acc = {0.f, 0.f, 0.f, 0.f, 0.f, 0.f, 0.f, 0.f};
    for (int k = 0; k < K; k += 4) {
      const float* ap = A + (size_t)mr * lda + k + half * 2;
      v2f a; a.x = ap[0]; a.y = ap[1];
      int kr = k + half * 2;
      v2f b; b.x = sW[kr * 16 + (lane & 15)];
             b.y = sW[(kr + 1) * 16 + (lane & 15)];
      acc = __builtin_amdgcn_wmma_f32_16x16x4_f32(false, a, false, b,
                                                  (short)0, acc, false, false);
    }
    float bc = bias[col];
#pragma unroll
    for (int r = 0; r < 8; ++r) {
      int m = row0 + r + half * 8;
      if (m < M) C[(size_t)m * ldc + col] = acc[r] + bc;
    }
  }
}

// ---------------------------------------------------------------------------
// Softmax state init
// ---------------------------------------------------------------------------
__global__ void init_softmax(unsigned* __restrict__ mEnc, float* __restrict__ denom, int n) {
  int i = blockIdx.x * blockDim.x + threadIdx.x;
  if (i < n) { mEnc[i] = 0u; denom[i] = 0.f; }
}

// ---------------------------------------------------------------------------
// Edge pass 1: logits[e,h] = q[dst]·(k[src]+We·ea) / sqrt(C); atomicMax per dst
// qkvs layout: [N, 4D], sections q|k|v|agg(skip), D = 8*C
// ---------------------------------------------------------------------------
template <int C>
__global__ void edge_logits_max(const int* __restrict__ src, const int* __restrict__ dst,
                                const float* __restrict__ ea, const float* __restrict__ We,
                                const float* __restrict__ qkvs, int ld,
                                float* __restrict__ logits, unsigned* __restrict__ mEnc, int E) {
  constexpr int D = 8 * C;
  __shared__ float sWe[7 * D];
  for (int i = threadIdx.x; i < 7 * D; i += blockDim.x) sWe[i] = We[i];
  __syncthreads();
  int idx = blockIdx.x * blockDim.x + threadIdx.x;
  if (idx >= E * 8) return;
  int e = idx >> 3, h = idx & 7;
  int s = src[e], d = dst[e];
  float ev[7];
#pragma unroll
  for (int j = 0; j < 7; ++j) ev[j] = ea[(size_t)e * 7 + j];
  const float* q = qkvs + (size_t)d * ld + h * C;
  const float* k = qkvs + (size_t)s * ld + D + h * C;
  float acc = 0.f;
#pragma unroll
  for (int c = 0; c < C; ++c) {
    float ee = 0.f;
#pragma unroll
    for (int j = 0; j < 7; ++j) ee += ev[j] * sWe[j * D + h * C + c];
    acc += q[c] * (k[c] + ee);
  }
  acc *= (1.0f / __builtin_sqrtf((float)C));
  logits[idx] = acc;
  atomicMax(&mEnc[d * 8 + h], encF(acc));
}

// ---------------------------------------------------------------------------
// Edge pass 2: ex = exp(logit - max); denom += ex  (logits overwritten with ex)
// ---------------------------------------------------------------------------
__global__ void edge_exp_sum(const int* __restrict__ dst, float* __restrict__ logits,
                             const unsigned* __restrict__ mEnc, float* __restrict__ denom, int E) {
  int idx = blockIdx.x * blockDim.x + threadIdx.x;
  if (idx >= E * 8) return;
  int e = idx >> 3, h = idx & 7;
  int d = dst[e];
  float ex = expf(logits[idx] - decF(mEnc[d * 8 + h]));
  logits[idx] = ex;
  atomicAdd(&denom[d * 8 + h], ex);
}

// ---------------------------------------------------------------------------
// Edge pass 3: agg[dst] += alpha * (v[src] + We·ea); agg is the skip section
// ---------------------------------------------------------------------------
template <int C>
__global__ void edge_aggregate(const int* __restrict__ src, const int* __restrict__ dst,
                               const float* __restrict__ ea, const float* __restrict__ We,
                               const float* __restrict__ logits, const float* __restrict__ denom,
                               float* __restrict__ qkvs, int ld, int E) {
  constexpr int D = 8 * C;
  __shared__ float sWe[7 * D];
  for (int i = threadIdx.x; i < 7 * D; i += blockDim.x) sWe[i] = We[i];
  __syncthreads();
  int idx = blockIdx.x * blockDim.x + threadIdx.x;
  if (idx >= E * 8) return;
  int e = idx >> 3, h = idx & 7;
  int s = src[e], d = dst[e];
  float alpha = logits[idx] / denom[d * 8 + h];
  float ev[7];
#pragma unroll
  for (int j = 0; j < 7; ++j) ev[j] = ea[(size_t)e * 7 + j];
  const float* v = qkvs + (size_t)s * ld + 2 * D + h * C;
  float* agg = qkvs + (size_t)d * ld + 3 * D + h * C;
#pragma unroll
  for (int c = 0; c < C; ++c) {
    float ee = 0.f;
#pragma unroll
    for (int j = 0; j < 7; ++j) ee += ev[j] * sWe[j * D + h * C + c];
    atomicAdd(&agg[c], alpha * (v[c] + ee));
  }
}

// ---------------------------------------------------------------------------
// In-place strided elu over a [rows, cols] section with leading dim ld
// ---------------------------------------------------------------------------
__global__ void elu_strided(float* __restrict__ base, int rows, int cols, int ld) {
  int i = blockIdx.x * blockDim.x + threadIdx.x;
  if (i >= rows * cols) return;
  int r = i / cols, c = i - r * cols;
  float* p = base + (size_t)r * ld + c;
  *p = eluf(*p);
}

// ---------------------------------------------------------------------------
// Head: out = elu(elu(h2) @ lin1 + b1) @ lin2 + b2 ; h2 = qkvs2[:,192:256]
// ---------------------------------------------------------------------------
__global__ void head_kernel(const float* __restrict__ qkvs2,
                            const float* __restrict__ l1w, const float* __restrict__ l1b,
                            const float* __restrict__ l2w, const float* __restrict__ l2b,
                            float* __restrict__ out, int N) {
  __shared__ float s1[64 * 20], sb1[20], s2[20], sb2;
  for (int i = threadIdx.x; i < 64 * 20; i += blockDim.x) s1[i] = l1w[i];
  for (int i = threadIdx.x; i < 20; i += blockDim.x) { sb1[i] = l1b[i]; s2[i] = l2w[i]; }
  if (threadIdx.x == 0) sb2 = l2b[0];
  __syncthreads();
  int n = blockIdx.x * blockDim.x + threadIdx.x;
  if (n >= N) return;
  float h[64];
#pragma unroll
  for (int c = 0; c < 64; ++c) h[c] = eluf(qkvs2[(size_t)n * 256 + 192 + c]);
  float t[20];
  for (int o = 0; o < 20; ++o) {
    float acc = sb1[o];
#pragma unroll
    for (int c = 0; c < 64; ++c) acc += h[c] * s1[c * 20 + o];
    t[o] = eluf(acc);
  }
  float acc = sb2;
#pragma unroll
  for (int o = 0; o < 20; ++o) acc += t[o] * s2[o];
  out[n] = acc;
}

// ---------------------------------------------------------------------------
extern "C" void kernel_launch(void* const* d_in, const int* in_sizes, int n_in,
                              void* d_out, int out_size, void* d_ws, size_t ws_size,
                              hipStream_t stream) {
  const float* x      = (const float*)d_in[0];
  const int*   eidx   = (const int*)  d_in[1];
  const float* ea     = (const float*)d_in[2];
  const float* linA_w = (const float*)d_in[3];
  const float* linA_b = (const float*)d_in[4];
  const float* linB_w = (const float*)d_in[5];
  const float* linB_b = (const float*)d_in[6];
  const float* q1_w = (const float*)d_in[7];  const float* q1_b = (const float*)d_in[8];
  const float* k1_w = (const float*)d_in[9];  const float* k1_b = (const float*)d_in[10];
  const float* v1_w = (const float*)d_in[11]; const float* v1_b = (const float*)d_in[12];
  const float* e1_w = (const float*)d_in[13];
  const float* s1_w = (const float*)d_in[14]; const float* s1_b = (const float*)d_in[15];
  const float* q2_w = (const float*)d_in[16]; const float* q2_b = (const float*)d_in[17];
  const float* k2_w = (const float*)d_in[18]; const float* k2_b = (const float*)d_in[19];
  const float* v2_w = (const float*)d_in[20]; const float* v2_b = (const float*)d_in[21];
  const float* e2_w = (const float*)d_in[22];
  const float* s2_w = (const float*)d_in[23]; const float* s2_b = (const float*)d_in[24];
  const float* l1_w = (const float*)d_in[25]; const float* l1_b = (const float*)d_in[26];
  const float* l2_w = (const float*)d_in[27]; const float* l2_b = (const float*)d_in[28];

  const int N = in_sizes[0] / 23;
  const int E = in_sizes[2] / 7;
  const int* src = eidx;
  const int* dstp = eidx + E;

  // workspace carve-out
  char* ws = (char*)d_ws;
  size_t off = 0;
  auto carve = [&](size_t bytes) -> void* {
    off = (off + 255) & ~(size_t)255;
    void* p = ws + off;
    off += bytes;
    return p;
  };
  float*    h0     = (float*)   carve((size_t)N * 24 * 4);
  float*    qkvs1  = (float*)   carve((size_t)N * 768 * 4);
  float*    qkvs2  = (float*)   carve((size_t)N * 256 * 4);
  float*    logits = (float*)   carve((size_t)E * 8 * 4);
  unsigned* mEnc   = (unsigned*)carve((size_t)N * 8 * 4);
  float*    denom  = (float*)   carve((size_t)N * 8 * 4);
  float*    W1p    = (float*)   carve((size_t)24 * 768 * 4);
  float*    b1p    = (float*)   carve(768 * 4);
  float*    W2p    = (float*)   carve((size_t)192 * 256 * 4);
  float*    b2p    = (float*)   carve(256 * 4);

  const int BT = 256;

  // weight packing
  {
    int t1 = 24 * 768 + 768;
    pack_w4<<<(t1 + BT - 1) / BT, BT, 0, stream>>>(q1_w, k1_w, v1_w, s1_w,
                                                   q1_b, k1_b, v1_b, s1_b,
                                                   W1p, b1p, 23, 24, 192);
    int t2 = 192 * 256 + 256;
    pack_w4<<<(t2 + BT - 1) / BT, BT, 0, stream>>>(q2_w, k2_w, v2_w, s2_w,
                                                   q2_b, k2_b, v2_b, s2_b,
                                                   W2p, b2p, 192, 192, 64);
  }

  // node MLP
  mlp_node<<<(N + BT - 1) / BT, BT, 0, stream>>>(x, linA_w, linA_b, linB_w, linB_b, h0, N);

  const int mtiles = (N + 15) / 16;
  const int mblocks = (mtiles + 7) / 8;
  const int nEH = E * 8;
  const int gEH = (nEH + BT - 1) / BT;
  const int gNH = (N * 8 + BT - 1) / BT;

  // ---- conv1 (heads=8, C=24, D=192) ----
  {
    wmma_gemm_f32<<<mblocks * (768 / 16), BT, 0, stream>>>(h0, 24, W1p, b1p,
                                                           qkvs1, 768, N, 24, 768);
    init_softmax<<<gNH, BT, 0, stream>>>(mEnc, denom, N * 8);
    edge_logits_max<24><<<gEH, BT, 0, stream>>>(src, dstp, ea, e1_w, qkvs1, 768, logits, mEnc, E);
    edge_exp_sum<<<gEH, BT, 0, stream>>>(dstp, logits, mEnc, denom, E);
    edge_aggregate<24><<<gEH, BT, 0, stream>>>(src, dstp, ea, e1_w, logits, denom, qkvs1, 768, E);
    int tl = N * 192;
    elu_strided<<<(tl + BT - 1) / BT, BT, 0, stream>>>(qkvs1 + 576, N, 192, 768);
  }

  // ---- conv2 (heads=8, C=8, D=64) ----
  {
    wmma_gemm_f32<<<mblocks * (256 / 16), BT, 0, stream>>>(qkvs1 + 576, 768, W2p, b2p,
                                                           qkvs2, 256, N, 192, 256);
    init_softmax<<<gNH, BT, 0, stream>>>(mEnc, denom, N * 8);
    edge_logits_max<8><<<gEH, BT, 0, stream>>>(src, dstp, ea, e2_w, qkvs2, 256, logits, mEnc, E);
    edge_exp_sum<<<gEH, BT, 0, stream>>>(dstp, logits, mEnc, denom, E);
    edge_aggregate<8><<<gEH, BT, 0, stream>>>(src, dstp, ea, e2_w, logits, denom, qkvs2, 256, E);
  }

  // head (applies elu to conv2 output internally)
  head_kernel<<<(N + BT - 1) / BT, BT, 0, stream>>>(qkvs2, l1_w, l1_b, l2_w, l2_b,
                                                    (float*)d_out, N);
}